// BahadanauAttention_45586782880446
// MI455X (gfx1250) — compile-verified
//
#include <hip/hip_runtime.h>

// ---------------------------------------------------------------------------
// Cosine-similarity attention context, MI455X (gfx1250, wave32).
//   dots    = keys @ q                      (v_wmma_f32_16x16x4_f32, exact f32)
//   k_norms = ||keys_row||                  (v_pk_mul_f32 + one shfl_xor(16))
//   score_s = dots_s / (||q|| * k_norms_s)  (v_readlane broadcast -> SGPR-ish)
//   out     = sum_s score_s * keys[s]       (coalesced float4 pass, L2-resident)
// Roofline: keys = 128 MB f32; one compulsory HBM read (~5.5 us @ 23.3 TB/s).
// The score-weighted second pass re-reads keys, which fit the 192 MB L2, so
// HBM traffic stays at the compulsory 128 MB. Arithmetic (~0.2 GFLOP) and the
// f32 WMMA stream both sit well under the memory time.
// ---------------------------------------------------------------------------

typedef float v2f __attribute__((ext_vector_type(2)));
typedef float v4f __attribute__((ext_vector_type(4)));
typedef float v8f __attribute__((ext_vector_type(8)));

#define H_DIM          1024
#define ROWS_PER_WAVE  16
#define WAVES_PER_BLK  8
#define BLOCK_THREADS  256   // 8 wave32

// uniform-lane broadcast: v_readlane_b32 (not ds_bpermute)
__device__ __forceinline__ float bcast_lane(float x, int srcLane) {
    return __int_as_float(__builtin_amdgcn_readlane(__float_as_int(x), srcLane));
}

__global__ __launch_bounds__(BLOCK_THREADS)
void cosattn_fused_kernel(const float* __restrict__ q,
                          const float* __restrict__ keys,
                          float* __restrict__ out,
                          int S)
{
    __shared__ float s_wsum[WAVES_PER_BLK];
    __shared__ float s_invqn;
    __shared__ float s_ctx[H_DIM];

    const int tid  = threadIdx.x;
    const int lane = tid & 31;
    const int widx = tid >> 5;

    // ---- 1/||q|| (block-redundant; 4 KB, L2-hot) + zero the LDS context ----
    {
        v4f q4 = *(const v4f*)(q + tid * 4);
        float qp = q4.x * q4.x + q4.y * q4.y + q4.z * q4.z + q4.w * q4.w;
        qp += __shfl_xor(qp, 16);
        qp += __shfl_xor(qp, 8);
        qp += __shfl_xor(qp, 4);
        qp += __shfl_xor(qp, 2);
        qp += __shfl_xor(qp, 1);
        if (lane == 0) s_wsum[widx] = qp;
    }
    *(v4f*)(s_ctx + tid * 4) = (v4f){0.f, 0.f, 0.f, 0.f};
    __syncthreads();
    if (tid == 0) {
        float s = 0.f;
        #pragma unroll
        for (int w = 0; w < WAVES_PER_BLK; ++w) s += s_wsum[w];
        s_invqn = __frsqrt_rn(s);
    }
    __syncthreads();
    const float inv_qn = s_invqn;

    // ---- per-wave 16-row tile ----
    const int  gw = blockIdx.x * WAVES_PER_BLK + widx;
    const long r0 = (long)gw * ROWS_PER_WAVE;
    const bool active = (r0 + ROWS_PER_WAVE) <= (long)S;   // wave-uniform

    float sc[ROWS_PER_WAVE];
    v4f   acc[8];
    #pragma unroll
    for (int j = 0; j < 8; ++j) acc[j] = (v4f){0.f, 0.f, 0.f, 0.f};

    if (active) {
        // -------- pass 1: dots via WMMA (A layout: lane=M, lane-half=K pair),
        //                  row sum-of-squares via VALU --------
        const int   hi   = lane >> 4;           // 0: K={0,1}, 1: K={2,3}
        const int   mrow = lane & 15;           // M index
        const float* arow = keys + (r0 + mrow) * H_DIM + hi * 2;
        const float* qrow = q + hi * 2;         // B[k,n] = q[c+k] for all n

        v8f   cd  = (v8f){0.f, 0.f, 0.f, 0.f, 0.f, 0.f, 0.f, 0.f};
        float nsq = 0.f;
        #pragma unroll 8
        for (int c = 0; c < H_DIM; c += 4) {
            v2f a  = *(const v2f*)(arow + c);
            v2f bq = *(const v2f*)(qrow + c);
            nsq += a.x * a.x;
            nsq += a.y * a.y;
            // D = A(16x4 keys tile) * B(4x16 q broadcast) + C  -> all 16
            // columns of row i equal dot(keys[r0+i], q) partial.
            cd = __builtin_amdgcn_wmma_f32_16x16x4_f32(
                     false, a, false, bq, (short)0, cd, false, false);
        }
        // row i total sumsq: lanes i and i+16 hold the two K-halves
        float nsq_tot = nsq + __shfl_xor(nsq, 16);

        // extract 16 dots (C/D layout: row i -> VGPR i%8, lanes 0-15 / 16-31)
        // via v_readlane broadcasts (indices are compile-time uniform)
        #pragma unroll
        for (int i = 0; i < ROWS_PER_WAVE; ++i) {
            float dot_i = bcast_lane(cd[i & 7], (i < 8) ? 0 : 16);
            float nrm_i = bcast_lane(nsq_tot, i & 15);
            sc[i] = dot_i * __frsqrt_rn(nrm_i) * inv_qn;
        }

        // -------- pass 2: score-weighted accumulate (re-read hits L2) --------
        for (int r = 0; r < ROWS_PER_WAVE; ++r) {
            const float* row = keys + (r0 + r) * H_DIM;
            if (r + 1 < ROWS_PER_WAVE)
                __builtin_prefetch(keys + (r0 + r + 1) * H_DIM + lane * 4, 0, 0);
            const float s = sc[r];
            #pragma unroll
            for (int j = 0; j < 8; ++j) {
                v4f v = *(const v4f*)(row + j * 128 + lane * 4);
                acc[j] += s * v;
            }
        }
    }

    // ---- combine: wave partials -> LDS (ds_add_f32) ----
    if (active) {
        #pragma unroll
        for (int j = 0; j < 8; ++j) {
            float* p = s_ctx + j * 128 + lane * 4;
            __hip_atomic_fetch_add(p + 0, acc[j].x, __ATOMIC_RELAXED, __HIP_MEMORY_SCOPE_WORKGROUP);
            __hip_atomic_fetch_add(p + 1, acc[j].y, __ATOMIC_RELAXED, __HIP_MEMORY_SCOPE_WORKGROUP);
            __hip_atomic_fetch_add(p + 2, acc[j].z, __ATOMIC_RELAXED, __HIP_MEMORY_SCOPE_WORKGROUP);
            __hip_atomic_fetch_add(p + 3, acc[j].w, __ATOMIC_RELAXED, __HIP_MEMORY_SCOPE_WORKGROUP);
        }
    }
    __syncthreads();

    // ---- block partial -> global (global_atomic_add_f32; 256 adds/addr) ----
    {
        v4f r = *(const v4f*)(s_ctx + tid * 4);
        float* p = out + tid * 4;
        __hip_atomic_fetch_add(p + 0, r.x, __ATOMIC_RELAXED, __HIP_MEMORY_SCOPE_AGENT);
        __hip_atomic_fetch_add(p + 1, r.y, __ATOMIC_RELAXED, __HIP_MEMORY_SCOPE_AGENT);
        __hip_atomic_fetch_add(p + 2, r.z, __ATOMIC_RELAXED, __HIP_MEMORY_SCOPE_AGENT);
        __hip_atomic_fetch_add(p + 3, r.w, __ATOMIC_RELAXED, __HIP_MEMORY_SCOPE_AGENT);
    }
}

extern "C" void kernel_launch(void* const* d_in, const int* in_sizes, int n_in,
                              void* d_out, int out_size, void* d_ws, size_t ws_size,
                              hipStream_t stream)
{
    (void)n_in; (void)d_ws; (void)ws_size;
    const float* q    = (const float*)d_in[0];   // [1,1024] f32
    const float* keys = (const float*)d_in[1];   // [32768,1024] f32
    float* out        = (float*)d_out;           // [1,1024] f32

    const int S = in_sizes[1] / H_DIM;           // 32768

    // output is accumulated with atomics; zero it first (graph-capturable)
    hipMemsetAsync(d_out, 0, (size_t)out_size * sizeof(float), stream);

    const int rows_per_block = WAVES_PER_BLK * ROWS_PER_WAVE;  // 128
    const int blocks = (S + rows_per_block - 1) / rows_per_block;  // 256
    cosattn_fused_kernel<<<blocks, BLOCK_THREADS, 0, stream>>>(q, keys, out, S);
}